// SharedSharedLoss_12790412607987
// MI455X (gfx1250) — compile-verified
//
#include <hip/hip_runtime.h>
#include <hip/hip_bf16.h>

typedef __attribute__((ext_vector_type(2))) float v2f;
typedef __attribute__((ext_vector_type(8))) float v8f;

#define D          2048      // feature dim
#define NVIEWS     4
#define ROWS_PER_G 16        // 4 chunks of 4 views per group -> 16x16 WMMA gram
#define NGROUPS    512       // 8192 rows / 16
#define WAVES_PER_BLOCK 4
#define KSLICE     (D / WAVES_PER_BLOCK)   // 512 K per wave
#define KTILE      64
#define MARGIN     0.3f
#define EPS_NORM   1e-12f
#define EPS_DIST   1e-12f

// One block (128 threads = 4 waves) per group of 16 rows.
// Each wave accumulates a partial 16x16 Gram over its K slice using
// v_wmma_f32_16x16x4_f32 with A==B (Gram trick), partials are summed in LDS,
// then 16 lanes run the 4x4 distance epilogue. Block result -> d_ws[group].
__global__ __launch_bounds__(WAVES_PER_BLOCK * 32)
void gram_loss_kernel(const float* __restrict__ X, float* __restrict__ ws) {
    const int g      = blockIdx.x;              // group id, 0..511
    const int lane   = threadIdx.x & 31;
    const int waveId = threadIdx.x >> 5;
    const int half   = (lane >> 4) & 1;         // 0: K pair {0,1}, 1: {2,3}
    const int mrow   = lane & 15;               // row within group this lane feeds to A

    const float* rowPtr = X + (size_t)(g * ROWS_PER_G + mrow) * D;
    const int k0 = waveId * KSLICE;

    v8f acc = {};   // partial 16x16 Gram (f32 C/D layout)

    for (int kt = k0; kt < k0 + KSLICE; kt += KTILE) {
        // speculative prefetch one tile ahead (dropped silently if OOB)
        __builtin_prefetch(rowPtr + kt + KTILE + 2 * half, 0, 0);
#pragma unroll
        for (int k = kt; k < kt + KTILE; k += 4) {
            // A layout 16x4 f32: lanes 0-15 hold K={k,k+1}, lanes 16-31 K={k+2,k+3}
            v2f a = *reinterpret_cast<const v2f*>(rowPtr + k + 2 * half);
            // Gram: B registers == A registers computes X * X^T
            acc = __builtin_amdgcn_wmma_f32_16x16x4_f32(
                false, a, false, a, (short)0, acc, false, false);
        }
    }

    // ---- cross-wave reduction of partial Grams ----
    __shared__ float sPart[WAVES_PER_BLOCK * 256];
    __shared__ float sG[ROWS_PER_G][ROWS_PER_G + 1];
    __shared__ float sRow[ROWS_PER_G];

#pragma unroll
    for (int r = 0; r < 8; ++r)
        sPart[waveId * 256 + r * 32 + lane] = acc[r];
    __syncthreads();

    if (waveId == 0) {
        // C/D layout: VGPR r holds M = r + 8*half, N = lane&15
#pragma unroll
        for (int r = 0; r < 8; ++r) {
            float t = sPart[0 * 256 + r * 32 + lane]
                    + sPart[1 * 256 + r * 32 + lane]
                    + sPart[2 * 256 + r * 32 + lane]
                    + sPart[3 * 256 + r * 32 + lane];
            sG[r + 8 * half][mrow] = t;   // mrow == N here (column index)
        }
    }
    __syncthreads();

    // ---- epilogue: per-row max off-diagonal distance within its chunk ----
    if (threadIdx.x < ROWS_PER_G) {
        const int i  = threadIdx.x;
        const int c0 = (i >> 2) << 2;             // first row of this chunk
        const float Gii = sG[i][i];
        const float mi  = fmaxf(sqrtf(Gii), EPS_NORM);
        const float sqi = Gii / (mi * mi);
        float maxd = -3.4e38f;
#pragma unroll
        for (int jj = 0; jj < NVIEWS; ++jj) {
            const int j = c0 + jj;
            if (j == i) continue;
            const float Gjj = sG[j][j];
            const float mj  = fmaxf(sqrtf(Gjj), EPS_NORM);
            const float sqj = Gjj / (mj * mj);
            const float d2  = sqi + sqj - 2.0f * sG[i][j] / (mi * mj);
            const float dst = sqrtf(fmaxf(d2, EPS_DIST));
            maxd = fmaxf(maxd, dst);
        }
        // mean over 4 views == sum/4 -> 0.25 per row
        sRow[i] = fmaxf(maxd + MARGIN, 0.0f) * 0.25f;
    }
    __syncthreads();

    if (threadIdx.x == 0) {
        float s = 0.0f;
#pragma unroll
        for (int i = 0; i < ROWS_PER_G; ++i) s += sRow[i];
        ws[g] = s;
    }
}

// Deterministic fixed-order tree reduction of the 512 per-group partials.
__global__ __launch_bounds__(NGROUPS)
void reduce_kernel(const float* __restrict__ ws, float* __restrict__ out) {
    __shared__ float s[NGROUPS];
    const int t = threadIdx.x;
    s[t] = ws[t];
    __syncthreads();
    for (int stride = NGROUPS / 2; stride > 0; stride >>= 1) {
        if (t < stride) s[t] += s[t + stride];
        __syncthreads();
    }
    if (t == 0) out[0] = s[0];
}

extern "C" void kernel_launch(void* const* d_in, const int* in_sizes, int n_in,
                              void* d_out, int out_size, void* d_ws, size_t ws_size,
                              hipStream_t stream) {
    (void)in_sizes; (void)n_in; (void)out_size; (void)ws_size;
    const float* X = (const float*)d_in[0];
    float* out = (float*)d_out;
    float* ws  = (float*)d_ws;

    gram_loss_kernel<<<NGROUPS, WAVES_PER_BLOCK * 32, 0, stream>>>(X, ws);
    reduce_kernel<<<1, NGROUPS, 0, stream>>>(ws, out);
}